// Attention_InfLevel_24343874634352
// MI455X (gfx1250) — compile-verified
//
#include <hip/hip_runtime.h>

typedef float v2f __attribute__((ext_vector_type(2)));
typedef float v8f __attribute__((ext_vector_type(8)));
typedef int   v4i __attribute__((vector_size(16)));   // matches builtin proto

#define AS1 __attribute__((address_space(1)))
#define AS3 __attribute__((address_space(3)))

#define GAMMA_F      0.5f
#define NEG_SLOPE_F  0.2f

#if __has_builtin(__builtin_amdgcn_global_load_async_to_lds_b128) && \
    __has_builtin(__builtin_amdgcn_s_wait_asynccnt)
#define USE_ASYNC_LDS 1
#else
#define USE_ASYNC_LDS 0
#endif

// ---------------------------------------------------------------------------
// Kernel 1: fused row dots.  rows [0,N) -> e1 = input1 . a1
//                            rows [N,N+M) -> e2 = input2 . a2
// ---------------------------------------------------------------------------
__global__ __launch_bounds__(128) void gat_rowdot_kernel(
    const float* __restrict__ input1, const float* __restrict__ input2,
    const float* __restrict__ a1, const float* __restrict__ a2,
    float* __restrict__ e_out, int N, int M, int D)
{
    int wave = threadIdx.x >> 5;
    int lane = threadIdx.x & 31;
    int r = blockIdx.x * 4 + wave;
    if (r >= N + M) return;
    const float* row;
    const float* a;
    if (r < N) { row = input1 + (size_t)r * D;        a = a1; }
    else       { row = input2 + (size_t)(r - N) * D;  a = a2; }
    float s = 0.0f;
    for (int k = lane; k < D; k += 32) s += row[k] * a[k];
    for (int off = 16; off > 0; off >>= 1) s += __shfl_xor(s, off, 32);
    if (lane == 0) e_out[r] = s;
}

// ---------------------------------------------------------------------------
// Kernel 2: per-row masked-softmax stats, single float4-streamed pass of adj.
// rowscale[i] = GAMMA * deg_i / Z_i (0 for empty rows); rowmax[i] = max_i.
// Non-edges contribute exactly 0 to Z (exp(-9e15 - max) underflows in f32).
// ---------------------------------------------------------------------------
__global__ __launch_bounds__(256) void gat_rowstats_kernel(
    const float* __restrict__ adj, const float* __restrict__ e1,
    const float* __restrict__ e2,
    float* __restrict__ rowmax, float* __restrict__ rowscale, int M)
{
    __shared__ float red[256];
    int i   = blockIdx.x;
    int tid = threadIdx.x;
    const float4* arow4 = (const float4*)(adj + (size_t)i * M);
    float e1i = e1[i];

    float ev[32];
    int cnt4 = M >> 10;               // M / (256*4)  (== 8)
    float lmax = -__builtin_inff();
    float ldeg = 0.0f;
    for (int k = 0; k < cnt4; ++k) {
        int v = tid + (k << 8);       // coalesced float4 stream
        float4 a4 = arow4[v];
        #pragma unroll
        for (int c = 0; c < 4; ++c) {
            float a = (&a4.x)[c];
            float e = -__builtin_inff();
            if (a > 0.0f) {
                int j = v * 4 + c;
                e = e1i + e2[j];
                e = (e > 0.0f) ? e : NEG_SLOPE_F * e;   // leaky_relu
                ldeg += a;
                lmax = fmaxf(lmax, e);
            }
            ev[k * 4 + c] = e;
        }
    }

    red[tid] = lmax; __syncthreads();
    for (int s = 128; s > 0; s >>= 1) {
        if (tid < s) red[tid] = fmaxf(red[tid], red[tid + s]);
        __syncthreads();
    }
    float m = red[0]; __syncthreads();

    red[tid] = ldeg; __syncthreads();
    for (int s = 128; s > 0; s >>= 1) {
        if (tid < s) red[tid] += red[tid + s];
        __syncthreads();
    }
    float deg = red[0]; __syncthreads();

    float mf = (deg > 0.0f) ? m : 0.0f;   // avoid (-inf)-(-inf)
    float lsum = 0.0f;
    for (int k = 0; k < 32; ++k) lsum += expf(ev[k] - mf);  // exp(-inf)=0

    red[tid] = lsum; __syncthreads();
    for (int s = 128; s > 0; s >>= 1) {
        if (tid < s) red[tid] += red[tid + s];
        __syncthreads();
    }
    if (tid == 0) {
        float Z = red[0];
        rowmax[i]   = mf;
        rowscale[i] = (deg > 0.0f && Z > 0.0f) ? (GAMMA_F * deg / Z) : 0.0f;
    }
}

// ---------------------------------------------------------------------------
// Kernel 3: out[32 rows x 256 cols per block] = W(32 x M) * input2(M x 256)
// via V_WMMA_F32_16X16X4_F32. 512 threads = 16 waves; wave w: row-tile w>>3,
// column pair (w&7)*32. One staged 16x256 input2 tile is shared by both
// row-tiles (halves L2 traffic vs 16-row blocks).
//   W on the fly: adj>0 -> scale_i*exp(lrelu(e1_i+e2_j)-max_i) + (1-GAMMA)
// A tile stored K-pair-interleaved in LDS -> A fragment = one ds_load_b64.
// input2 tile staged double-buffered via GLOBAL_LOAD_ASYNC_TO_LDS_B128
// (each wave issues exactly 2/iter; s_wait_asynccnt(2) before the barrier
// guarantees the current buffer is complete for all waves). Register-float4
// fallback if the async builtins are unavailable.
// ---------------------------------------------------------------------------
__global__ __launch_bounds__(512) void gat_attn_wmma_kernel(
    const float* __restrict__ adj, const float* __restrict__ input2,
    const float* __restrict__ e1, const float* __restrict__ e2,
    const float* __restrict__ rowmax, const float* __restrict__ rowscale,
    float* __restrict__ out, int M)
{
    constexpr int D    = 256;
    constexpr int BPAD = D + 8;  // row stride 264 (==8 mod 64 banks; the two
                                 // lane halves of a B-frag use disjoint banks)
    __shared__ __align__(16) float sB[2][16 * BPAD];     // input2 tile x2
    // w^T, K-pair interleaved: [pair p=0..7][m=0..31][k&1], row stride 48
    // (pair-to-pair = 96 floats == 32 mod 64 banks -> conflict-free halves)
    __shared__ __align__(16) float sWT[8 * 48 * 2];

    int tid  = threadIdx.x;
    int lane = tid & 31;
    int wave = tid >> 5;
    int i0   = blockIdx.x * 32;

    int rt = wave >> 3;          // row-tile 0/1 (rows i0+16*rt ..)
    int c0 = (wave & 7) * 32;    // this wave's first output column

    // weight-tile construction mapping (adj loads: 16 consecutive tids read
    // 64 contiguous bytes of one adj row)
    int i_loc = tid >> 4;        // 0..31
    int j_loc = tid & 15;
    size_t adj_row = (size_t)(i0 + i_loc) * M;
    float m_i = rowmax  [i0 + i_loc];
    float s_i = rowscale[i0 + i_loc];
    float e1i = e1      [i0 + i_loc];
    int   wt_idx = (((j_loc >> 1) * 48) + i_loc) * 2 + (j_loc & 1);

    bool hi = lane >= 16;
    int  l  = lane & 15;

    v8f acc0 = {};
    v8f acc1 = {};

    const int NB = M >> 4;       // 16-wide K blocks

    // ---- pipeline prologue ----
    float a_cur  = adj[adj_row + j_loc];
    float e2_cur = e2[j_loc];
#if USE_ASYNC_LDS
    {   // issue async stage of buffer 0 (each wave issues exactly 2 b128s)
        #pragma unroll
        for (int v = 0; v < 2; ++v) {
            int idx = tid + v * 512;           // 1024 float4 chunks
            int row = idx >> 6;                // 64 float4 per row
            int c4  = idx & 63;
            AS1 v4i* gsrc = (AS1 v4i*)(input2 + (size_t)row * D + c4 * 4);
            AS3 v4i* ldst = (AS3 v4i*)(&sB[0][row * BPAD + c4 * 4]);
            __builtin_amdgcn_global_load_async_to_lds_b128(gsrc, ldst, 0, 0);
        }
    }
#else
    float4 bc[2];
    {
        const float4* src = (const float4*)input2;
        #pragma unroll
        for (int v = 0; v < 2; ++v) {
            int idx = tid + v * 512;
            int row = idx >> 6;
            int c4  = idx & 63;
            bc[v] = src[row * (D / 4) + c4];
        }
    }
#endif

    for (int jb = 0; jb < NB; ++jb) {
        int j0  = jb << 4;
        int cur = jb & 1;
        int jn  = (jb + 1 < NB) ? (j0 + 16) : 0;   // wrap: harmless dummy

        // ---- build transposed weight tile from pipelined regs ----
        {
            float w = 0.0f;
            if (a_cur > 0.0f) {
                float e = e1i + e2_cur;
                e = (e > 0.0f) ? e : NEG_SLOPE_F * e;
                w = s_i * expf(e - m_i) + (1.0f - GAMMA_F);
            }
            sWT[wt_idx] = w;
        }

#if USE_ASYNC_LDS
        // issue next tile's async copies into the other buffer
        #pragma unroll
        for (int v = 0; v < 2; ++v) {
            int idx = tid + v * 512;
            int row = idx >> 6;
            int c4  = idx & 63;
            AS1 v4i* gsrc =
                (AS1 v4i*)(input2 + (size_t)(jn + row) * D + c4 * 4);
            AS3 v4i* ldst = (AS3 v4i*)(&sB[cur ^ 1][row * BPAD + c4 * 4]);
            __builtin_amdgcn_global_load_async_to_lds_b128(gsrc, ldst, 0, 0);
        }
        // wait until only the 2 just-issued copies are outstanding:
        // current buffer is then complete (ASYNCcnt completes in order)
        __builtin_amdgcn_s_wait_asynccnt(2);
#else
        {   // store pipelined registers into the (single) staging buffer
            #pragma unroll
            for (int v = 0; v < 2; ++v) {
                int idx = tid + v * 512;
                int row = idx >> 6;
                int c4  = idx & 63;
                *(float4*)(&sB[0][row * BPAD + c4 * 4]) = bc[v];
            }
        }
#endif
        __syncthreads();

        // ---- preload next tile's adj / e2 (overlaps the WMMAs) ----
        a_cur  = adj[adj_row + jn + j_loc];
        e2_cur = e2[jn + j_loc];
#if !USE_ASYNC_LDS
        {
            const float4* src = (const float4*)(input2 + (size_t)jn * D);
            #pragma unroll
            for (int v = 0; v < 2; ++v) {
                int idx = tid + v * 512;
                int row = idx >> 6;
                int c4  = idx & 63;
                bc[v] = src[row * (D / 4) + c4];
            }
        }
#endif

        // ---- 16-deep K block = 4 x (2 n-tiles) WMMA f32 16x16x4 ----
        const float* B = &sB[USE_ASYNC_LDS ? cur : 0][0];
        #pragma unroll
        for (int k0 = 0; k0 < 16; k0 += 4) {
            int pa = (k0 >> 1) + (hi ? 1 : 0);   // A/B lane-half K offset
            int ka = k0 + (hi ? 2 : 0);
            v2f af = *(const v2f*)&sWT[(pa * 48 + rt * 16 + l) * 2];
            v2f b0, b1;
            b0.x = B[ka * BPAD + c0 + l];
            b0.y = B[(ka + 1) * BPAD + c0 + l];
            b1.x = B[ka * BPAD + c0 + 16 + l];
            b1.y = B[(ka + 1) * BPAD + c0 + 16 + l];
            acc0 = __builtin_amdgcn_wmma_f32_16x16x4_f32(
                       false, af, false, b0, (short)0, acc0, false, false);
            acc1 = __builtin_amdgcn_wmma_f32_16x16x4_f32(
                       false, af, false, b1, (short)0, acc1, false, false);
        }
        __syncthreads();
    }

    // ---- write back: VGPR r -> row r (lanes 0-15) / r+8 (lanes 16-31) ----
    int rbase = i0 + rt * 16 + (hi ? 8 : 0);
    #pragma unroll
    for (int r = 0; r < 8; ++r) {
        out[(size_t)(rbase + r) * D + c0 + l]      = acc0[r];
        out[(size_t)(rbase + r) * D + c0 + 16 + l] = acc1[r];
    }
}

// ---------------------------------------------------------------------------
// Launch
// ---------------------------------------------------------------------------
extern "C" void kernel_launch(void* const* d_in, const int* in_sizes, int n_in,
                              void* d_out, int out_size, void* d_ws, size_t ws_size,
                              hipStream_t stream)
{
    const float* input1 = (const float*)d_in[0];   // (N, D)
    const float* input2 = (const float*)d_in[1];   // (M, D)
    const float* adj    = (const float*)d_in[2];   // (N, M)
    const float* a1     = (const float*)d_in[3];   // (D, 1)
    const float* a2     = (const float*)d_in[4];   // (D, 1)
    float* out = (float*)d_out;                    // (N, D)

    int D = in_sizes[3];              // 256
    int N = in_sizes[0] / D;          // 8192
    int M = in_sizes[1] / D;          // 8192

    float* ws     = (float*)d_ws;
    float* e1     = ws;               // [N]
    float* e2     = ws + N;           // [M]
    float* rmax   = ws + N + M;       // [N]
    float* rscale = ws + N + M + N;   // [N]

    gat_rowdot_kernel<<<(N + M + 3) / 4, 128, 0, stream>>>(
        input1, input2, a1, a2, ws, N, M, D);
    gat_rowstats_kernel<<<N, 256, 0, stream>>>(
        adj, e1, e2, rmax, rscale, M);
    gat_attn_wmma_kernel<<<N / 32, 512, 0, stream>>>(
        adj, input2, e1, e2, rmax, rscale, out, M);
}